// OCAE_26456998544138
// MI455X (gfx1250) — compile-verified
//
#include <hip/hip_runtime.h>
#include <math.h>

#define B_TOT 16384
#define K 24
#define SF 16
#define OPW 26            // SF + 1 + 9
#define EPS 1e-6f
#define TWO_PI_CUBED 248.05021344239853f   // (2*pi)^3

typedef __attribute__((ext_vector_type(2))) float v2f;
typedef __attribute__((ext_vector_type(8))) float v8f;

__global__ void ocae_zero_kernel(float* out_ll) {
    if (threadIdx.x == 0 && blockIdx.x == 0) out_ll[0] = 0.0f;
}

__global__ __launch_bounds__(256)
void ocae_main_kernel(const float* __restrict__ OP,   // [B,K,26]
                      const float* __restrict__ XM,   // [B,K,6]
                      const float* __restrict__ DM,   // [B,K]
                      const float* __restrict__ N1,   // [B,K]
                      const float* __restrict__ N2,   // [B,K]
                      const float* __restrict__ W1,   // [K,16,16]
                      const float* __restrict__ B1,   // [K,16]
                      const float* __restrict__ W2,   // [K,16,48]
                      const float* __restrict__ B2,   // [K,48]
                      const float* __restrict__ OPM,  // [K,K,3,3]
                      float* __restrict__ out_ll,     // [1]
                      float* __restrict__ out_ak,     // [B,K]
                      float* __restrict__ out_akn,    // [B,K,K]
                      float* __restrict__ out_g)      // [B,K,K] (lambda parked here, then gaussian)
{
    __shared__ float lds_h[8][16 * 17];       // per-wave h transpose buffer
    __shared__ float lds_ov[16 * K * 9];      // relu'd poses
    __shared__ float lds_ak[16 * K];
    __shared__ float lds_aknsum[16 * K];
    __shared__ float lds_aksum[16];
    __shared__ float lds_mix[16 * K];
    __shared__ float lds_ll;

    const int tid  = threadIdx.x;
    const int lane = tid & 31;
    const int wave = tid >> 5;
    const int m    = lane & 15;   // matmul row (batch) for A-loads, column for B/C
    const int hi   = lane >> 4;   // lane-half selector
    const int r0   = blockIdx.x * 16;

    // ================= Phase A: per-capsule MLPs via V_WMMA_F32_16X16X4_F32 ========
    for (int cap = 0; cap < 3; ++cap) {
        const int k = wave * 3 + cap;

        // ---- matmul1: relu(features)[16x16] @ W1[k][16x16] ----
        v8f acc = {0.f, 0.f, 0.f, 0.f, 0.f, 0.f, 0.f, 0.f};
        const size_t arow = ((size_t)(r0 + m) * K + k) * OPW + 9;
        for (int t = 0; t < 4; ++t) {
            const int kc = 4 * t + 2 * hi;
            v2f a, b;
            a.x = fmaxf(OP[arow + kc],     0.0f);
            a.y = fmaxf(OP[arow + kc + 1], 0.0f);
            b.x = W1[(k * SF + kc)     * SF + m];
            b.y = W1[(k * SF + kc + 1) * SF + m];
            acc = __builtin_amdgcn_wmma_f32_16x16x4_f32(false, a, false, b,
                                                        (short)0, acc, false, false);
        }
        const float b1v = B1[k * SF + m];

        __syncthreads();   // prior iteration's lds_h reads fully done / ordering fence
        #pragma unroll
        for (int j = 0; j < 8; ++j)
            lds_h[wave][(j + 8 * hi) * 17 + m] = fmaxf(acc[j] + b1v, 0.0f);
        __syncthreads();   // h visible for transposed A-reads

        // ---- matmul2: h[16x16] @ W2[k][16x48], three 16-wide N tiles ----
        for (int nt = 0; nt < 3; ++nt) {
            const float b2v = B2[k * 48 + nt * 16 + m];
            v8f acc2;
            #pragma unroll
            for (int j = 0; j < 8; ++j) acc2[j] = b2v;   // bias (per column) pre-loaded

            for (int t = 0; t < 4; ++t) {
                const int kc = 4 * t + 2 * hi;
                v2f a, b;
                a.x = lds_h[wave][m * 17 + kc];          // h in A layout (row=m, K=kc)
                a.y = lds_h[wave][m * 17 + kc + 1];
                b.x = W2[(k * SF + kc)     * 48 + nt * 16 + m];
                b.y = W2[(k * SF + kc + 1) * 48 + nt * 16 + m];
                acc2 = __builtin_amdgcn_wmma_f32_16x16x4_f32(false, a, false, b,
                                                             (short)0, acc2, false, false);
            }

            const int ng = nt * 16 + m;                  // global output column 0..47
            #pragma unroll
            for (int j = 0; j < 8; ++j) {
                const int rloc = j + 8 * hi;
                const size_t rowg = (size_t)(r0 + rloc);
                const float val = acc2[j];
                if (ng < K) {
                    // a_kn = sigmoid(mlp[...,:24] + noise_2[b, n])
                    const float akn = 1.0f / (1.0f + expf(-(val + N2[rowg * K + ng])));
                    out_akn[(rowg * K + k) * K + ng] = akn;
                } else {
                    // lambda = relu(mlp[...,24:]) + eps  -> parked in gaussian slot
                    const float lam = fmaxf(val, 0.0f) + EPS;
                    out_g[(rowg * K + k) * K + (ng - K)] = lam;
                }
            }
        }
    }

    __threadfence();   // publish a_kn / lambda globals to all waves of this block
    __syncthreads();

    // ================= Phase B prologue: a_k, poses, row sums ======================
    for (int idx = tid; idx < 16 * K; idx += 256) {
        const int row = idx / K, kk = idx % K;
        const size_t rowg = (size_t)(r0 + row);
        const float ak = 1.0f / (1.0f + expf(-(OP[(rowg * K + kk) * OPW + 25] +
                                               N1[rowg * K + kk])));
        lds_ak[idx] = ak;
        out_ak[rowg * K + kk] = ak;
        float s = 0.0f;
        const size_t base = (rowg * K + kk) * K;
        for (int n = 0; n < K; ++n) s += out_akn[base + n];
        lds_aknsum[idx] = s;
        lds_mix[idx] = (float)K * EPS;   // sum over k of gauss_mix includes K*EPS
    }
    if (tid == 0) lds_ll = 0.0f;
    for (int idx = tid; idx < 16 * K * 9; idx += 256) {
        const int row = idx / (K * 9);
        const int rem = idx % (K * 9);
        const int kk = rem / 9, i = rem % 9;
        lds_ov[idx] = fmaxf(OP[((size_t)(r0 + row) * K + kk) * OPW + i], 0.0f);
    }
    __syncthreads();
    for (int idx = tid; idx < 16; idx += 256) {
        float s = 0.0f;
        for (int kk = 0; kk < K; ++kk) s += lds_ak[idx * K + kk];
        lds_aksum[idx] = s;
    }
    __syncthreads();

    // ================= Phase B main: gaussian + mixture (16*24*24 per block) =======
    for (int idx = tid; idx < 16 * K * K; idx += 256) {
        const int row = idx / (K * K);
        const int kn  = idx % (K * K);
        const int k = kn / K, n = kn % K;
        const size_t rowg = (size_t)(r0 + row);

        const float* ov = &lds_ov[(row * K + k) * 9];
        const float* Mp = &OPM[((size_t)k * K + n) * 9];
        const float* xp = &XM[(rowg * K + n) * 6];

        float ss = 0.0f;
        #pragma unroll
        for (int c = 0; c < 6; ++c) {
            const int i = c / 3, l = c % 3;
            const float mu = ov[3 * i] * Mp[l] + ov[3 * i + 1] * Mp[3 + l] +
                             ov[3 * i + 2] * Mp[6 + l];
            const float d = xp[c] - mu;
            ss += d * d;
        }

        const size_t gi = (rowg * K + k) * K + n;
        const float lam = out_g[gi];                         // parked lambda
        const float g = expf(-0.5f * ss / lam) / (TWO_PI_CUBED * lam * lam * lam);
        out_g[gi] = g;                                        // overwrite with gaussian

        const float akn = out_akn[gi];
        const float gc = lds_ak[row * K + k] * akn /
                         (lds_aksum[row] * lds_aknsum[row * K + n]);
        atomicAdd(&lds_mix[row * K + n], g * gc);
    }
    __syncthreads();

    // ================= Epilogue: log, weight by d_m, reduce ========================
    float part = 0.0f;
    for (int idx = tid; idx < 16 * K; idx += 256) {
        const int row = idx / K, n = idx % K;
        part += logf(lds_mix[idx]) * DM[(size_t)(r0 + row) * K + n];
    }
    atomicAdd(&lds_ll, part);
    __syncthreads();
    if (tid == 0) atomicAdd(out_ll, lds_ll * (1.0f / (float)B_TOT));
}

extern "C" void kernel_launch(void* const* d_in, const int* in_sizes, int n_in,
                              void* d_out, int out_size, void* d_ws, size_t ws_size,
                              hipStream_t stream) {
    const float* OP  = (const float*)d_in[0];
    const float* XM  = (const float*)d_in[1];
    const float* DM  = (const float*)d_in[2];
    const float* N1  = (const float*)d_in[3];
    const float* N2  = (const float*)d_in[4];
    const float* W1  = (const float*)d_in[5];
    const float* B1  = (const float*)d_in[6];
    const float* W2  = (const float*)d_in[7];
    const float* B2  = (const float*)d_in[8];
    const float* OPM = (const float*)d_in[9];

    float* out = (float*)d_out;
    float* out_ll  = out;                                             // [1]
    float* out_ak  = out + 1;                                         // [B,K]
    float* out_akn = out + 1 + (size_t)B_TOT * K;                     // [B,K,K]
    float* out_g   = out + 1 + (size_t)B_TOT * K + (size_t)B_TOT * K * K; // [B,K,K]

    ocae_zero_kernel<<<1, 64, 0, stream>>>(out_ll);
    ocae_main_kernel<<<B_TOT / 16, 256, 0, stream>>>(OP, XM, DM, N1, N2,
                                                     W1, B1, W2, B2, OPM,
                                                     out_ll, out_ak, out_akn, out_g);
}